// GraphTransformerRefiner_3453153706682
// MI455X (gfx1250) — compile-verified
//
#include <hip/hip_runtime.h>
#include <stdint.h>

// ---------------------------------------------------------------------------
// Graph transformer layer for MI455X (gfx1250, wave32).
//   h    = x @ W_in + b_in
//   q/k/v/skip = h @ W* + b*
//   score[e,h] = <q[dst],k[src]>_head / sqrt(32); softmax over incoming edges
//   agg[dst]  += alpha * v[src]
//   out  = (agg + skip) @ W_out + b_out
// GEMMs run on V_WMMA_F32_16X16X4_F32 (fp32 matrix pipe), weights staged in
// LDS with a pair-swizzled layout so B fragments are single ds_load_b64s.
// ---------------------------------------------------------------------------

#define N_NODES 50000
#define IN_DIM  128
#define HC      128        // HEADS * OUT_DIM
#define OUT_DIM 32
#define HEADS   4
#define N_EDGES 800000

typedef float v2f __attribute__((ext_vector_type(2)));
typedef float v8f __attribute__((ext_vector_type(8)));

// ---------------------------------------------------------------------------
// Generic row-major GEMM: Y[nrows, P] = (X (+ Xadd)) [nrows,128] @ W[128,P] + b
// P = NT*16. Block = 256 threads = 8 waves; wave w owns a 16-row strip
// (blockIdx.x*8 + w) and computes all NT column tiles (A-fragment reuse).
//
// LDS weight layout (pair-swizzled for b64 B-fragment loads):
//   sW[(k>>1)*(2P) + 2n + (k&1)] = W[k][n]
// so lane's B pair {W[k][n], W[k+1][n]} (k even) is contiguous 8B.
//
// fp32 WMMA fragment layouts (ISA 7.12.2, 16x16x4 f32):
//   A (16x4): lane holds A[m = lane&15][k = 2*(lane>>4) + j] in vector elem j
//   B (4x16): lane holds B[k = 2*(lane>>4) + j][n = lane&15] in vector elem j
//   C (16x16): vector elem r holds C[m = r + 8*(lane>>4)][n = lane&15]
// ---------------------------------------------------------------------------
template <int NT, bool HAS_ADD>
__global__ __launch_bounds__(256) void gemm_wmma_f32(
    const float* __restrict__ X,     // [nrows, 128]
    const float* __restrict__ Xadd,  // elementwise add operand (HAS_ADD only)
    const float* __restrict__ W,     // [128, P] row-major
    const float* __restrict__ bias,  // [P]
    float* __restrict__ Y,           // [nrows, P]
    int nrows) {
  constexpr int P = NT * 16;
  extern __shared__ float sW[];  // 128 * P floats, pair-swizzled

  for (int i = threadIdx.x; i < 128 * P; i += 256) {
    const int k = i / P;
    const int n = i - k * P;
    sW[(k >> 1) * (2 * P) + 2 * n + (k & 1)] = W[i];
  }
  __syncthreads();

  const int wave = threadIdx.x >> 5;
  const int lane = threadIdx.x & 31;
  const int row0 = (blockIdx.x * 8 + wave) * 16;
  if (row0 >= nrows) return;  // wave-uniform: EXEC stays all-1s for WMMA

  const int m = lane & 15;       // M (A) / N (B,C) index within tile
  const int khalf = lane >> 4;   // selects K pair {0,1} or {2,3}

  const float* xrow = X + (size_t)(row0 + m) * 128;
  const float* xrow2 = Xadd + (size_t)(row0 + m) * 128;  // unused if !HAS_ADD

  // Per-lane base into swizzled LDS: pair row (k0/2 + khalf), column pair 2m.
  const float* sb = sW + khalf * (2 * P) + 2 * m;

  v8f acc[NT] = {};

  for (int k0 = 0; k0 < 128; k0 += 4) {
    const int k = k0 + 2 * khalf;  // even -> 8B-aligned
    v2f a = *(const v2f*)(xrow + k);
    if (HAS_ADD) {
      v2f a2 = *(const v2f*)(xrow2 + k);
      a += a2;
    }
    const float* sbk = sb + (k0 >> 1) * (2 * P);
#pragma unroll
    for (int t = 0; t < NT; ++t) {
      v2f b = *(const v2f*)(sbk + 2 * 16 * t);  // single ds_load_b64
      acc[t] = __builtin_amdgcn_wmma_f32_16x16x4_f32(
          /*neg_a=*/false, a, /*neg_b=*/false, b,
          /*c_mod=*/(short)0, acc[t], /*reuse_a=*/false, /*reuse_b=*/false);
    }
  }

#pragma unroll
  for (int t = 0; t < NT; ++t) {
    const int n = t * 16 + m;
    const float bn = bias[n];
#pragma unroll
    for (int r = 0; r < 8; ++r) {
      const int mr = r + 8 * khalf;
      Y[(size_t)(row0 + mr) * P + n] = acc[t][r] + bn;
    }
  }
}

// ---------------------------------------------------------------------------
__global__ void zero_f32(float* __restrict__ p, size_t n) {
  size_t i = (size_t)blockIdx.x * blockDim.x + threadIdx.x;
  if (i < n) p[i] = 0.0f;
}

// One thread per (edge, head): score = <q[dst], k[src]>_head / sqrt(C);
// ex = exp(score) (max-subtraction omitted: identical softmax, scores tiny);
// denom[dst,head] accumulated with L2 float atomics.
__global__ __launch_bounds__(256) void edge_scores(
    const float* __restrict__ q, const float* __restrict__ k,
    const int64_t* __restrict__ srcI, const int64_t* __restrict__ dstI,
    float* __restrict__ ex, float* __restrict__ denom, int nedge) {
  int idx = blockIdx.x * blockDim.x + threadIdx.x;  // edge*HEADS + head
  if (idx >= nedge * HEADS) return;
  const int e = idx >> 2;
  const int h = idx & 3;
  const int64_t s = srcI[e], d = dstI[e];
  const float4* qp = (const float4*)(q + (size_t)d * HC + h * OUT_DIM);
  const float4* kp = (const float4*)(k + (size_t)s * HC + h * OUT_DIM);
  float acc = 0.0f;
#pragma unroll
  for (int i = 0; i < OUT_DIM / 4; ++i) {
    float4 a = qp[i], b = kp[i];
    acc += a.x * b.x + a.y * b.y + a.z * b.z + a.w * b.w;
  }
  const float ev = expf(acc * 0.17677669529663687f);  // 1/sqrt(32)
  ex[idx] = ev;
  atomicAdd(&denom[(size_t)d * HEADS + h], ev);
}

// One wave per edge, lane owns 4 channels (head = lane>>3):
// agg[dst, c] += alpha(e, head) * v[src, c]
__global__ __launch_bounds__(256) void edge_aggregate(
    const float* __restrict__ v, const int64_t* __restrict__ srcI,
    const int64_t* __restrict__ dstI, const float* __restrict__ ex,
    const float* __restrict__ denom, float* __restrict__ agg, int nedge) {
  const int e = blockIdx.x * (blockDim.x >> 5) + (threadIdx.x >> 5);
  if (e >= nedge) return;
  const int lane = threadIdx.x & 31;
  const int h = lane >> 3;
  const int64_t s = srcI[e], d = dstI[e];
  const float alpha =
      ex[(size_t)e * HEADS + h] / denom[(size_t)d * HEADS + h];
  const float4 vv = *(const float4*)(v + (size_t)s * HC + lane * 4);
  float* out = agg + (size_t)d * HC + lane * 4;
  atomicAdd(out + 0, vv.x * alpha);
  atomicAdd(out + 1, vv.y * alpha);
  atomicAdd(out + 2, vv.z * alpha);
  atomicAdd(out + 3, vv.w * alpha);
}

// ---------------------------------------------------------------------------
extern "C" void kernel_launch(void* const* d_in, const int* in_sizes, int n_in,
                              void* d_out, int out_size, void* d_ws,
                              size_t ws_size, hipStream_t stream) {
  (void)in_sizes; (void)n_in; (void)out_size; (void)ws_size;

  const float*   x      = (const float*)d_in[0];
  const int64_t* eidx   = (const int64_t*)d_in[1];  // [2, E] int64
  const float*   W_in   = (const float*)d_in[2];
  const float*   b_in   = (const float*)d_in[3];
  const float*   Wq     = (const float*)d_in[4];
  const float*   bq     = (const float*)d_in[5];
  const float*   Wk     = (const float*)d_in[6];
  const float*   bk     = (const float*)d_in[7];
  const float*   Wv     = (const float*)d_in[8];
  const float*   bv     = (const float*)d_in[9];
  const float*   Wskip  = (const float*)d_in[10];
  const float*   bskip  = (const float*)d_in[11];
  const float*   W_out  = (const float*)d_in[12];
  const float*   b_out  = (const float*)d_in[13];
  float*         out    = (float*)d_out;

  const int64_t* srcI = eidx;            // edge_index[0]
  const int64_t* dstI = eidx + N_EDGES;  // edge_index[1]

  // Workspace layout (fp32). agg aliases h: h is dead after q/k/v/skip GEMMs.
  const size_t nodeF = (size_t)N_NODES * HC;  // 6.4M floats
  float* h     = (float*)d_ws;
  float* q     = h + nodeF;
  float* k     = q + nodeF;
  float* v     = k + nodeF;
  float* skip  = v + nodeF;
  float* ex    = skip + nodeF;                      // E*HEADS
  float* denom = ex + (size_t)N_EDGES * HEADS;      // N*HEADS
  float* agg   = h;                                 // alias (h dead by then)

  const int rowTiles = N_NODES / 16;                 // 3125 (exact)
  const dim3 gemmGrid((rowTiles + 7) / 8);           // 8 waves * 16 rows/block
  const dim3 blk256(256);
  const size_t shW128 = 128 * 128 * sizeof(float);   // 64 KB
  const size_t shW32  = 128 * 32 * sizeof(float);    // 16 KB

  // 1) h = x @ W_in + b_in
  gemm_wmma_f32<8, false><<<gemmGrid, blk256, shW128, stream>>>(
      x, nullptr, W_in, b_in, h, N_NODES);
  // 2) q/k/v/skip projections from h
  gemm_wmma_f32<8, false><<<gemmGrid, blk256, shW128, stream>>>(
      h, nullptr, Wq, bq, q, N_NODES);
  gemm_wmma_f32<8, false><<<gemmGrid, blk256, shW128, stream>>>(
      h, nullptr, Wk, bk, k, N_NODES);
  gemm_wmma_f32<8, false><<<gemmGrid, blk256, shW128, stream>>>(
      h, nullptr, Wv, bv, v, N_NODES);
  gemm_wmma_f32<8, false><<<gemmGrid, blk256, shW128, stream>>>(
      h, nullptr, Wskip, bskip, skip, N_NODES);

  // 3) zero accumulators every call (graph replay safe; h is dead now)
  {
    const size_t nAgg = nodeF;
    const size_t nDen = (size_t)N_NODES * HEADS;
    zero_f32<<<dim3((unsigned)((nAgg + 255) / 256)), blk256, 0, stream>>>(agg, nAgg);
    zero_f32<<<dim3((unsigned)((nDen + 255) / 256)), blk256, 0, stream>>>(denom, nDen);
  }

  // 4) per-edge attention logits -> exp + softmax denominators
  {
    const int total = N_EDGES * HEADS;
    edge_scores<<<dim3((total + 255) / 256), blk256, 0, stream>>>(
        q, k, srcI, dstI, ex, denom, N_EDGES);
  }

  // 5) weighted scatter of v into agg (one wave per edge)
  edge_aggregate<<<dim3(N_EDGES / 8), blk256, 0, stream>>>(
      v, srcI, dstI, ex, denom, agg, N_EDGES);

  // 6) out = (agg + skip) @ W_out + b_out
  gemm_wmma_f32<2, true><<<gemmGrid, blk256, shW32, stream>>>(
      agg, skip, W_out, b_out, out, N_NODES);
}